// AdaptiveAdjacency_84035330113919
// MI455X (gfx1250) — compile-verified
//
#include <hip/hip_runtime.h>
#include <hip/hip_bf16.h>
#include <math.h>

#define N_NODES 16384
#define DIM     64
#define TOPK    20

typedef __attribute__((ext_vector_type(2))) float v2f;
typedef __attribute__((ext_vector_type(8))) float v8f;

// ---------------------------------------------------------------------------
// Kernel 1: L2-normalize rows.  One wave (32 lanes) per row; each lane owns a
// float2 of the 64-wide row.  norm = x * 1/sqrt(max(sum(x^2), 1e-12)).
// ---------------------------------------------------------------------------
__global__ __launch_bounds__(256) void l2norm_kernel(const float* __restrict__ emb,
                                                     float* __restrict__ nemb) {
    const int wave = threadIdx.x >> 5;
    const int lane = threadIdx.x & 31;
    const int row  = blockIdx.x * 8 + wave;   // 8 waves per block
    if (row >= N_NODES) return;

    const float2* rp = (const float2*)(emb + (size_t)row * DIM);
    float2 d = rp[lane];
    float s = d.x * d.x + d.y * d.y;
    // wave32 reduction
    #pragma unroll
    for (int off = 16; off > 0; off >>= 1)
        s += __shfl_xor(s, off, 32);
    float inv = 1.0f / sqrtf(fmaxf(s, 1e-12f));
    float2 o; o.x = d.x * inv; o.y = d.y * inv;
    ((float2*)(nemb + (size_t)row * DIM))[lane] = o;
}

// ---------------------------------------------------------------------------
// Kernel 2: streaming cosine-sim top-k via V_WMMA_F32_16X16X4_F32.
// Block = 128 threads = 4 waves.  Wave w owns rows [blk*64 + w*16, +16).
// Loop over all 1024 column tiles of 16; per tile: 16 chained f32 WMMAs
// (K = 64), transpose C through LDS, stream candidates into per-row top-20.
// ---------------------------------------------------------------------------
__global__ __launch_bounds__(128) void topk_sim_kernel(const float* __restrict__ nemb,
                                                       float* __restrict__ out) {
    const int lane = threadIdx.x & 31;
    const int wave = threadIdx.x >> 5;
    const int rowTileBase = blockIdx.x * 64 + wave * 16;

    __shared__ __align__(16) float Btile[16][DIM + 4];   // +4 floats pad: conflict-free
    __shared__ float simT[4][16][17];
    __shared__ float topV[4][16][TOPK];
    __shared__ int   topI[4][16][TOPK];

    // --- init per-row top lists (each owning lane touches only its region) ---
    if (lane < 16) {
        #pragma unroll
        for (int k = 0; k < TOPK; ++k) {
            topV[wave][lane][k] = -3.402823e38f;
            topI[wave][lane][k] = -1;
        }
    }
    float thr   = -3.402823e38f;  // current min of my top list (lanes 0-15)
    int   minPos = 0;

    // --- load A fragments for this wave's 16 rows, all K=64, kept resident ---
    // ISA A layout (16x4 f32): lane l -> M = l%16, elem j -> K = 2*(l>>4)+j
    v2f afrag[16];
    {
        const int m  = lane & 15;
        const int kb = (lane >> 4) * 2;
        const float* rowp = nemb + (size_t)(rowTileBase + m) * DIM;
        #pragma unroll
        for (int kk = 0; kk < 16; ++kk) {
            afrag[kk].x = rowp[kk * 4 + kb + 0];
            afrag[kk].y = rowp[kk * 4 + kb + 1];
        }
    }

    const int ldRow = threadIdx.x >> 3;   // 16 rows, 8 threads each
    const int ldSeg = threadIdx.x & 7;    // 2 float4 per thread

    for (int nt = 0; nt < N_NODES / 16; ++nt) {
        __syncthreads();  // prior iteration's Btile/simT reads are done

        // cooperative stage of B rows (columns nt*16 .. nt*16+15 of sim)
        {
            const float4* src = (const float4*)(nemb + ((size_t)nt * 16 + ldRow) * DIM);
            float4* dst = (float4*)(&Btile[ldRow][0]);
            dst[ldSeg]     = src[ldSeg];
            dst[ldSeg + 8] = src[ldSeg + 8];
        }
        __syncthreads();  // Btile ready

        // --- 16x16 tile of sim via 16 chained f32 WMMAs over K=64 ---
        // ISA B layout (4x16 f32): lane l -> N = l%16, elem j -> K = 2*(l>>4)+j
        v8f c = {0.0f, 0.0f, 0.0f, 0.0f, 0.0f, 0.0f, 0.0f, 0.0f};
        {
            const int n  = lane & 15;
            const int kb = (lane >> 4) * 2;
            #pragma unroll
            for (int kk = 0; kk < 16; ++kk) {
                v2f b;
                b.x = Btile[n][kk * 4 + kb + 0];
                b.y = Btile[n][kk * 4 + kb + 1];
                c = __builtin_amdgcn_wmma_f32_16x16x4_f32(
                        /*neg_a=*/false, afrag[kk],
                        /*neg_b=*/false, b,
                        /*c_mod=*/(short)0, c,
                        /*reuse_a=*/false, /*reuse_b=*/false);
            }
        }

        // --- transpose C through LDS: VGPR v -> row v + 8*(lane>>4), col lane%16
        {
            const int h = lane >> 4;
            const int n = lane & 15;
            #pragma unroll
            for (int v = 0; v < 8; ++v)
                simT[wave][v + 8 * h][n] = c[v];
        }
        __syncthreads();  // simT ready

        // --- stream this tile's 16 candidates into my row's top-20 ---
        if (lane < 16) {
            const int r = lane;
            #pragma unroll
            for (int j = 0; j < 16; ++j) {
                float val = simT[wave][r][j];
                if (val > thr) {
                    topV[wave][r][minPos] = val;
                    topI[wave][r][minPos] = nt * 16 + j;
                    float m = topV[wave][r][0];
                    int  mp = 0;
                    #pragma unroll
                    for (int k = 1; k < TOPK; ++k) {
                        float tv = topV[wave][r][k];
                        if (tv < m) { m = tv; mp = k; }
                    }
                    thr = m; minPos = mp;
                }
            }
        }
    }

    // --- final: sort 20 entries desc (tie -> lower index), sigmoid, self-mask ---
    if (lane < 16) {
        const int r = lane;
        const int globalRow = rowTileBase + r;
        float v[TOPK]; int id[TOPK];
        #pragma unroll
        for (int k = 0; k < TOPK; ++k) { v[k] = topV[wave][r][k]; id[k] = topI[wave][r][k]; }
        for (int i = 1; i < TOPK; ++i) {
            float vv = v[i]; int ii = id[i]; int j = i - 1;
            while (j >= 0 && (v[j] < vv || (v[j] == vv && id[j] > ii))) {
                v[j + 1] = v[j]; id[j + 1] = id[j]; --j;
            }
            v[j + 1] = vv; id[j + 1] = ii;
        }
        float* op = out + (size_t)globalRow * TOPK;
        #pragma unroll
        for (int k = 0; k < TOPK; ++k) {
            float s = 1.0f / (1.0f + expf(-v[k]));
            op[k] = (id[k] == globalRow) ? 0.0f : s;
        }
    }
}

// ---------------------------------------------------------------------------
extern "C" void kernel_launch(void* const* d_in, const int* in_sizes, int n_in,
                              void* d_out, int out_size, void* d_ws, size_t ws_size,
                              hipStream_t stream) {
    (void)in_sizes; (void)n_in; (void)out_size; (void)ws_size;
    const float* emb  = (const float*)d_in[0];
    float*       nemb = (float*)d_ws;          // 16384*64*4 = 4 MB scratch
    float*       outp = (float*)d_out;         // 16384*20 floats

    l2norm_kernel<<<N_NODES / 8, 256, 0, stream>>>(emb, nemb);
    topk_sim_kernel<<<N_NODES / 64, 128, 0, stream>>>(nemb, outp);
}